// CRF_39702677684372
// MI455X (gfx1250) — compile-verified
//
#include <hip/hip_runtime.h>
#include <hip/hip_bf16.h>

typedef float v2f __attribute__((ext_vector_type(2)));
typedef float v4f __attribute__((ext_vector_type(4)));
typedef float v8f __attribute__((ext_vector_type(8)));

#define T_LEN 16384
#define D_LEN 2048
#define K_TAG 16
#define CHUNK 128
#define NCHUNK 128          // ceil((T_LEN-1)/CHUNK)
#define NEG_INF (-1e30f)

#define KC 128              // k-columns staged per stage
#define NSTAGE (D_LEN / KC) // 16
#define LDW 132             // LDS row stride (132 mod 64 == 4 -> conflict-free frags)

// ---------------------------------------------------------------------------
// Kernel 1: emission GEMM  emit[T,16] = seq[T,2048] @ W.T + b
// 16x16 output tile per block (4 waves). seq/W staged through double-buffered
// LDS with fully coalesced per-row b128 global loads; one ds_load_b128 per
// lane feeds TWO V_WMMA_F32_16X16X4_F32 via a k-axis permutation (lanes 0-15
// own d base+{0..3}, lanes 16-31 own d base+{4..7}; .xy -> chunk A, .zw ->
// chunk B — valid since A and B share the same contraction permutation).
// ---------------------------------------------------------------------------
__global__ __launch_bounds__(128) void crf_emit_kernel(
    const float* __restrict__ seq, const float* __restrict__ W,
    const float* __restrict__ bias, float* __restrict__ emit) {
  __shared__ float As[2][16][LDW];
  __shared__ float Ws[2][16][LDW];

  const int tid  = threadIdx.x;
  const int lane = tid & 31;
  const int wv   = tid >> 5;            // wave 0..3
  const int tile = blockIdx.x;          // 0..T/16-1
  const int m    = lane & 15;           // A row / B col index
  const int k4   = (lane >> 4) * 4;     // 0 (lanes 0-15) or 4 (lanes 16-31)

  const float* seqT = seq + (size_t)tile * 16 * D_LEN;

  v4f ra[4], rw[4];

  // ---- preload stage 0 (coalesced: one wave = one contiguous 512B row run)
#pragma unroll
  for (int i = 0; i < 4; ++i) {
    const int sidx = tid + 128 * i;         // 512 b128 segments
    const int row  = sidx >> 5;
    const int c4   = (sidx & 31) * 4;
    ra[i] = *(const v4f*)(seqT + (size_t)row * D_LEN + c4);
    rw[i] = *(const v4f*)(W    + (size_t)row * D_LEN + c4);
  }
#pragma unroll
  for (int i = 0; i < 4; ++i) {
    const int sidx = tid + 128 * i;
    const int row  = sidx >> 5;
    const int c4   = (sidx & 31) * 4;
    *(v4f*)&As[0][row][c4] = ra[i];
    *(v4f*)&Ws[0][row][c4] = rw[i];
  }
  __syncthreads();

  v8f c = {};
  int buf = 0;
  for (int s = 0; s < NSTAGE; ++s) {
    // ---- prefetch next stage into registers (overlaps with compute below)
    if (s + 1 < NSTAGE) {
      const int base = (s + 1) * KC;
#pragma unroll
      for (int i = 0; i < 4; ++i) {
        const int sidx = tid + 128 * i;
        const int row  = sidx >> 5;
        const int c4   = (sidx & 31) * 4;
        ra[i] = *(const v4f*)(seqT + (size_t)row * D_LEN + base + c4);
        rw[i] = *(const v4f*)(W    + (size_t)row * D_LEN + base + c4);
      }
    }
    // ---- compute: wave wv owns cols [wv*32, wv*32+32) of this stage
#pragma unroll
    for (int p = 0; p < 4; ++p) {
      const int c8 = wv * 32 + p * 8;
      v4f a4 = *(const v4f*)&As[buf][m][c8 + k4];
      v4f b4 = *(const v4f*)&Ws[buf][m][c8 + k4];
      v2f aA = { a4.x, a4.y }, bA = { b4.x, b4.y };
      v2f aB = { a4.z, a4.w }, bB = { b4.z, b4.w };
      c = __builtin_amdgcn_wmma_f32_16x16x4_f32(false, aA, false, bA,
                                                (short)0, c, false, false);
      c = __builtin_amdgcn_wmma_f32_16x16x4_f32(false, aB, false, bB,
                                                (short)0, c, false, false);
    }
    // ---- commit prefetched stage to the other buffer
    if (s + 1 < NSTAGE) {
#pragma unroll
      for (int i = 0; i < 4; ++i) {
        const int sidx = tid + 128 * i;
        const int row  = sidx >> 5;
        const int c4   = (sidx & 31) * 4;
        *(v4f*)&As[buf ^ 1][row][c4] = ra[i];
        *(v4f*)&Ws[buf ^ 1][row][c4] = rw[i];
      }
    }
    __syncthreads();
    buf ^= 1;
  }

  // ---- reduce the 4 per-wave partial C tiles (reuse As as scratch)
  float (*part)[8][32] = (float (*)[8][32])As;
#pragma unroll
  for (int r = 0; r < 8; ++r) part[wv][r][lane] = c[r];
  __syncthreads();

  // C/D layout: VGPR r, lane L -> row = 8*(L>>4)+r, col = L&15
  for (int idx = tid; idx < 256; idx += 128) {
    const int r = idx >> 5, L = idx & 31;
    const float ssum =
        part[0][r][L] + part[1][r][L] + part[2][r][L] + part[3][r][L];
    const int row = tile * 16 + 8 * (L >> 4) + r;
    const int col = L & 15;
    emit[(size_t)row * K_TAG + col] = ssum + bias[col];
  }
}

// ---------------------------------------------------------------------------
// Kernel 2: per-chunk log-semiring matrix product (parallelized forward scan).
// ---------------------------------------------------------------------------
__global__ __launch_bounds__(256) void crf_chunk_kernel(
    const float* __restrict__ emit, const float* __restrict__ trans,
    float* __restrict__ chunkM) {
  __shared__ float tr[16][17];
  __shared__ float mat[2][16][17];
  const int tid = threadIdx.x;
  const int cur = tid >> 4, j = tid & 15;

  tr[cur][j] = trans[tid];                        // trans[cur][prev]
  mat[0][cur][j] = (cur == j) ? 0.0f : NEG_INF;   // log-semiring identity
  __syncthreads();

  int t0 = 1 + blockIdx.x * CHUNK;
  int t1 = t0 + CHUNK;
  if (t1 > T_LEN) t1 = T_LEN;

  int buf = 0;
  for (int t = t0; t < t1; ++t) {
    const float e = emit[t * K_TAG + cur];
    float v[16];
    float mx = NEG_INF;
#pragma unroll
    for (int p = 0; p < 16; ++p) {
      v[p] = tr[cur][p] + mat[buf][p][j];
      mx = fmaxf(mx, v[p]);
    }
    float s = 0.0f;
#pragma unroll
    for (int p = 0; p < 16; ++p) s += __expf(v[p] - mx);
    mat[buf ^ 1][cur][j] = e + mx + __logf(s);
    __syncthreads();
    buf ^= 1;
  }
  chunkM[blockIdx.x * 256 + tid] = mat[buf][cur][j];
}

// ---------------------------------------------------------------------------
// Kernel 3: fold chunk matrices, apply alpha0/trans_end, gold path score,
// out[0] = gold - log_z. Single block.
// ---------------------------------------------------------------------------
__global__ __launch_bounds__(256) void crf_final_kernel(
    const float* __restrict__ emit, const int* __restrict__ tags,
    const float* __restrict__ trans_start, const float* __restrict__ trans,
    const float* __restrict__ trans_end, const float* __restrict__ chunkM,
    float* __restrict__ out) {
  __shared__ float P[2][16][17];
  __shared__ float red[256];
  __shared__ float lse[256];
  const int tid = threadIdx.x;
  const int cur = tid >> 4, j = tid & 15;

  // ---- gold path partial sum
  float g = 0.0f;
  for (int t = tid; t < T_LEN; t += 256) {
    const int tg = tags[t];
    g += emit[t * K_TAG + tg];
    if (t > 0) g += trans[tg * K_TAG + tags[t - 1]];
  }
  red[tid] = g;

  // ---- fold chunk matrices (left fold: P <- M_c (x) P)
  P[0][cur][j] = chunkM[tid];   // chunk 0
  __syncthreads();
  int buf = 0;
  for (int c = 1; c < NCHUNK; ++c) {
    const float* A = chunkM + c * 256 + cur * 16;   // A[cur][mid]
    float v[16];
    float mx = NEG_INF;
#pragma unroll
    for (int p = 0; p < 16; ++p) {
      v[p] = A[p] + P[buf][p][j];
      mx = fmaxf(mx, v[p]);
    }
    float s = 0.0f;
#pragma unroll
    for (int p = 0; p < 16; ++p) s += __expf(v[p] - mx);
    P[buf ^ 1][cur][j] = mx + __logf(s);
    __syncthreads();
    buf ^= 1;
  }

  // val[cur][j] = trans_end[cur] + P[cur][j] + alpha0[j]
  lse[tid] = trans_end[cur] + P[buf][cur][j] + trans_start[j] + emit[j];
  __syncthreads();

  if (tid == 0) {
    float mx = NEG_INF;
    for (int i = 0; i < 256; ++i) mx = fmaxf(mx, lse[i]);
    float s = 0.0f;
    for (int i = 0; i < 256; ++i) s += __expf(lse[i] - mx);
    const float log_z = mx + __logf(s);
    float gold = 0.0f;
    for (int i = 0; i < 256; ++i) gold += red[i];
    gold += trans_start[tags[0]] + trans_end[tags[T_LEN - 1]];
    out[0] = gold - log_z;
  }
}

extern "C" void kernel_launch(void* const* d_in, const int* in_sizes, int n_in,
                              void* d_out, int out_size, void* d_ws, size_t ws_size,
                              hipStream_t stream) {
  const float* seq         = (const float*)d_in[0];
  const int*   tags        = (const int*)  d_in[1];
  const float* W           = (const float*)d_in[2];
  const float* b           = (const float*)d_in[3];
  const float* trans_start = (const float*)d_in[4];
  const float* trans       = (const float*)d_in[5];
  const float* trans_end   = (const float*)d_in[6];
  float* out = (float*)d_out;

  float* emit   = (float*)d_ws;                        // T*16 floats = 1 MB
  float* chunkM = emit + (size_t)T_LEN * K_TAG;        // NCHUNK*256 floats

  crf_emit_kernel<<<T_LEN / 16, 128, 0, stream>>>(seq, W, b, emit);
  crf_chunk_kernel<<<NCHUNK, 256, 0, stream>>>(emit, trans, chunkM);
  crf_final_kernel<<<1, 256, 0, stream>>>(emit, tags, trans_start, trans,
                                          trans_end, chunkM, out);
}